// IterativeSequential2D_55155970015297
// MI455X (gfx1250) — compile-verified
//
#include <hip/hip_runtime.h>

// ---------------------------------------------------------------------------
// Block-tridiagonal iterative layer for MI455X (gfx1250, wave32, WMMA).
//   X : [4, 512, 2048] f32    W : [10, 2048, 2048] f32    b : [10, 2048] f32
//   2 iterations of  Y_i = relu( sum_j X_j @ W_k^T + b_k ),  tridiagonal pairs.
// fp32 via V_WMMA_F32_16X16X4_F32; double-buffered LDS with async global->LDS
// loads (GLOBAL_LOAD_ASYNC_TO_LDS_B128, tracked by ASYNCcnt).
// ---------------------------------------------------------------------------

typedef float v2f __attribute__((ext_vector_type(2)));
typedef float v8f __attribute__((ext_vector_type(8)));
typedef int   v4i __attribute__((vector_size(16)));   // matches builtin param type

#define BSZ      2048      // features per block
#define NBATCH   512
#define NBLK     4
#define TM       128       // workgroup tile M (batch rows)
#define TN       64        // workgroup tile N (out features)
#define KC       32        // K chunk staged in LDS
#define LSTR     36        // KC + 4 pad floats: conflict-free frag loads
#define NTHREADS 128       // 4 wave32, arranged 2 (M) x 2 (N)
#define NCHUNK   (BSZ / KC)   // 64 chunks per GEMM

static_assert(NCHUNK == 64, "chunk shift below assumes 64");

#if defined(__has_builtin)
#  if __has_builtin(__builtin_amdgcn_global_load_async_to_lds_b128) && \
      __has_builtin(__builtin_amdgcn_s_wait_asynccnt)
#    define USE_ASYNC_LDS 1
#  endif
#endif
#ifndef USE_ASYNC_LDS
#  define USE_ASYNC_LDS 0
#endif

#define AS1 __attribute__((address_space(1)))
#define AS3 __attribute__((address_space(3)))

// per chunk, per thread: A tile 128x32 -> 8 x b128, B tile 64x32 -> 4 x b128
#define ASYNC_PER_CHUNK 12

__global__ __launch_bounds__(NTHREADS)
void bsparse_iter_kernel(const float* __restrict__ X,
                         const float* __restrict__ W,
                         const float* __restrict__ Bv,
                         float* __restrict__ Y) {
    __shared__ float ldsA[2][TM * LSTR];   // X rows (M x K)
    __shared__ float ldsB[2][TN * LSTR];   // W rows (N x K) -> B[k][n] = W[n][k]

    const int i   = blockIdx.z;            // output block 0..3
    const int m0  = blockIdx.y * TM;
    const int n0  = blockIdx.x * TN;
    const int tid = threadIdx.x;
    const int lane = tid & 31;
    const int wave = tid >> 5;             // 0..3
    const int wm   = wave >> 1;            // 0..1 -> 64 rows per wave
    const int wn   = wave & 1;             // 0..1 -> 32 cols per wave
    const int lh   = lane >> 4;            // lane half (K select)
    const int ll   = lane & 15;            // M (A) / N (B,D) index

    // tridiagonal pattern, uniform scalars (no indexed const arrays):
    const int jlo    = (i == 0) ? 0 : i - 1;
    const int jhi    = (i == NBLK - 1) ? (NBLK - 1) : i + 1;
    const int np     = jhi - jlo + 1;                // 2 or 3
    const int kstart = (i == 0) ? 0 : 3 * i - 1;     // 0,2,5,8
    const int nchunks = np * NCHUNK;

    v8f acc[4][2];
#pragma unroll
    for (int mi = 0; mi < 4; ++mi)
#pragma unroll
        for (int ni = 0; ni < 2; ++ni)
            acc[mi][ni] = (v8f){0.f, 0.f, 0.f, 0.f, 0.f, 0.f, 0.f, 0.f};

    // global source pointers for flattened chunk index c = p*64 + ci
    auto chunk_src = [&](int c, const float*& a, const float*& b) {
        const int p  = c >> 6;
        const int e0 = (c & (NCHUNK - 1)) * KC;
        a = X + ((size_t)(jlo + p)    * NBATCH + m0) * BSZ + e0;
        b = W + ((size_t)(kstart + p) * BSZ    + n0) * BSZ + e0;
    };

    auto compute = [&](int buf) {
        const float* la = &ldsA[buf][0];
        const float* lb = &ldsB[buf][0];
#pragma unroll
        for (int kk = 0; kk < KC / 4; ++kk) {
            const int off = kk * 4 + 2 * lh;   // per-lane K offset (2 floats)
            v2f af[4], bf[2];
#pragma unroll
            for (int mi = 0; mi < 4; ++mi)
                af[mi] = *(const v2f*)&la[(wm * 64 + mi * 16 + ll) * LSTR + off];
#pragma unroll
            for (int ni = 0; ni < 2; ++ni)
                bf[ni] = *(const v2f*)&lb[(wn * 32 + ni * 16 + ll) * LSTR + off];
#pragma unroll
            for (int mi = 0; mi < 4; ++mi)
#pragma unroll
                for (int ni = 0; ni < 2; ++ni)
                    acc[mi][ni] = __builtin_amdgcn_wmma_f32_16x16x4_f32(
                        false, af[mi], false, bf[ni],
                        (short)0, acc[mi][ni], false, false);
        }
    };

#if USE_ASYNC_LDS
    // ---- gfx1250 async global->LDS path (ASYNCcnt) ----
    auto issue = [&](int c, int buf) {
        const float *a, *b;
        chunk_src(c, a, b);
        float* la = &ldsA[buf][0];
        float* lb = &ldsB[buf][0];
#pragma unroll
        for (int it = 0; it < 8; ++it) {       // A: 128 x 32
            const int idx = it * NTHREADS + tid;
            const int r = idx >> 3, cc = (idx & 7) << 2;
            __builtin_amdgcn_global_load_async_to_lds_b128(
                (AS1 v4i*)(a + (size_t)r * BSZ + cc),
                (AS3 v4i*)(la + r * LSTR + cc), 0, 0);
        }
#pragma unroll
        for (int it = 0; it < 4; ++it) {       // B: 64 x 32
            const int idx = it * NTHREADS + tid;
            const int r = idx >> 3, cc = (idx & 7) << 2;
            __builtin_amdgcn_global_load_async_to_lds_b128(
                (AS1 v4i*)(b + (size_t)r * BSZ + cc),
                (AS3 v4i*)(lb + r * LSTR + cc), 0, 0);
        }
    };

    issue(0, 0);
    for (int c = 0; c < nchunks; ++c) {
        const int cur = c & 1;
        if (c + 1 < nchunks) {
            issue(c + 1, cur ^ 1);             // prefetch next chunk
            // async loads complete in order: <=12 outstanding means chunk c done
            __builtin_amdgcn_s_wait_asynccnt(ASYNC_PER_CHUNK);
        } else {
            __builtin_amdgcn_s_wait_asynccnt(0);
        }
        __syncthreads();                        // all threads' tiles in place
        compute(cur);
        __syncthreads();                        // buf cur free for c+2 prefetch
    }
#else
    // ---- fallback: register-staged double buffer ----
    float4 ra[8], rb[4];
    auto gload = [&](int c) {
        const float *a, *b;
        chunk_src(c, a, b);
#pragma unroll
        for (int it = 0; it < 8; ++it) {
            const int idx = it * NTHREADS + tid;
            const int r = idx >> 3, cc = (idx & 7) << 2;
            ra[it] = *(const float4*)(a + (size_t)r * BSZ + cc);
        }
#pragma unroll
        for (int it = 0; it < 4; ++it) {
            const int idx = it * NTHREADS + tid;
            const int r = idx >> 3, cc = (idx & 7) << 2;
            rb[it] = *(const float4*)(b + (size_t)r * BSZ + cc);
        }
    };
    auto lstore = [&](int buf) {
#pragma unroll
        for (int it = 0; it < 8; ++it) {
            const int idx = it * NTHREADS + tid;
            const int r = idx >> 3, cc = (idx & 7) << 2;
            *(float4*)&ldsA[buf][r * LSTR + cc] = ra[it];
        }
#pragma unroll
        for (int it = 0; it < 4; ++it) {
            const int idx = it * NTHREADS + tid;
            const int r = idx >> 3, cc = (idx & 7) << 2;
            *(float4*)&ldsB[buf][r * LSTR + cc] = rb[it];
        }
    };

    gload(0);
    lstore(0);
    for (int c = 0; c < nchunks; ++c) {
        const int cur = c & 1;
        if (c + 1 < nchunks) gload(c + 1);      // overlap with compute
        __syncthreads();                        // buf cur visible
        compute(cur);
        __syncthreads();                        // readers of buf cur^1 done
        if (c + 1 < nchunks) lstore(cur ^ 1);
    }
#endif

    // epilogue: summed bias + ReLU, direct f32 stores
#pragma unroll
    for (int ni = 0; ni < 2; ++ni) {
        const int col = n0 + wn * 32 + ni * 16 + ll;
        float bias = 0.f;
        for (int p = 0; p < np; ++p)
            bias += Bv[(size_t)(kstart + p) * BSZ + col];
#pragma unroll
        for (int mi = 0; mi < 4; ++mi) {
#pragma unroll
            for (int v = 0; v < 8; ++v) {
                const int row = m0 + wm * 64 + mi * 16 + v + 8 * lh;
                const float val = acc[mi][ni][v] + bias;
                Y[((size_t)i * NBATCH + row) * BSZ + col] = fmaxf(val, 0.f);
            }
        }
    }
}

extern "C" void kernel_launch(void* const* d_in, const int* in_sizes, int n_in,
                              void* d_out, int out_size, void* d_ws, size_t ws_size,
                              hipStream_t stream) {
    (void)in_sizes; (void)n_in; (void)out_size; (void)ws_size;
    const float* X  = (const float*)d_in[0];
    const float* W  = (const float*)d_in[1];
    const float* Bv = (const float*)d_in[2];
    float* tmp = (float*)d_ws;    // 4*512*2048 f32 = 16.78 MB intermediate
    float* out = (float*)d_out;

    dim3 grid(BSZ / TN, NBATCH / TM, NBLK);   // 32 x 4 x 4 = 512 workgroups
    dim3 block(NTHREADS);                     // 4 wave32

    // iteration 1: X -> tmp ; iteration 2: tmp -> out
    bsparse_iter_kernel<<<grid, block, 0, stream>>>(X,   W, Bv, tmp);
    bsparse_iter_kernel<<<grid, block, 0, stream>>>(tmp, W, Bv, out);
}